// RPNProposalSelectorLayer_41678362640439
// MI455X (gfx1250) — compile-verified
//
#include <hip/hip_runtime.h>
#include <hip/hip_bf16.h>
#include <stdint.h>

#define N_TOTAL   (1 * 512 * 512 * 20)
#define ROUNDS    100
#define CLS_TH    0.7f
#define NMS_TH    0.5f
#define IOU_EPS   1e-9f
#define BT        256

// ---- CDNA5 async global->LDS path (probe via __has_builtin; fallback = direct loads)
#if defined(__has_builtin)
#if __has_builtin(__builtin_amdgcn_global_load_async_to_lds_b128) && \
    __has_builtin(__builtin_amdgcn_global_load_async_to_lds_b64)  && \
    __has_builtin(__builtin_amdgcn_s_wait_asynccnt)
#define USE_ASYNC_LDS 1
#endif
#endif

// Exact parameter types per hipcc diagnostic:
//   b128: (int __attribute__((vector_size(16))) AS1*, ... )
//   b64 : (int __attribute__((vector_size(8)))  AS1*, ... )
typedef int v4i_t __attribute__((vector_size(16)));
typedef int v2i_t __attribute__((vector_size(8)));

// Address-space bridging via integer casts:
//  - generic->LDS(3): flat LDS addresses truncate to the 32-bit LDS offset (ISA 10.2)
//  - generic->global(1): numerically identical; also drops const
#define GPTR4(p) ((__attribute__((address_space(1))) v4i_t*)(unsigned long long)(p))
#define GPTR2(p) ((__attribute__((address_space(1))) v2i_t*)(unsigned long long)(p))
#define LPTR4(p) ((__attribute__((address_space(3))) v4i_t*)(unsigned int)(unsigned long long)(p))
#define LPTR2(p) ((__attribute__((address_space(3))) v2i_t*)(unsigned int)(unsigned long long)(p))

__device__ __forceinline__ unsigned long long pack_key(float s, unsigned int idx) {
    return ((unsigned long long)__float_as_uint(s) << 32) |
           (unsigned long long)(0xFFFFFFFFu - idx);
}

__device__ __forceinline__ bool iou_suppress(float4 b, float4 sb, float sa) {
    float y1 = fmaxf(sb.x, b.x);
    float x1 = fmaxf(sb.y, b.y);
    float y2 = fminf(sb.z, b.z);
    float x2 = fminf(sb.w, b.w);
    float inter = fmaxf(y2 - y1, 0.0f) * fmaxf(x2 - x1, 0.0f);
    float ab = (b.z - b.x) * (b.w - b.y);
    float iou = inter / (sa + ab - inter + IOU_EPS);
    return iou > NMS_TH;
}

// ---------------- suppress: IoU(sel, all survivors) > 0.5 -> kill key -----------------
// (defined FIRST so the disasm snippet shows the async global->LDS path)
__global__ __launch_bounds__(BT) void rpn_suppress(const float* __restrict__ prop,
                                                   const unsigned long long* __restrict__ keys, int r,
                                                   unsigned long long* __restrict__ keyc,
                                                   const float4* __restrict__ bxc,
                                                   const int* __restrict__ m, int cap,
                                                   float* __restrict__ out) {
    unsigned long long key = keys[r];
    if (key == 0ull) return;  // no valid box this round: d_out slot stays zero (init)

    float s = __uint_as_float((unsigned int)(key >> 32));
    unsigned int sel_i = 0xFFFFFFFFu - (unsigned int)(key & 0xFFFFFFFFull);
    float4 sb = ((const float4*)prop)[sel_i];
    if (blockIdx.x == 0 && threadIdx.x == 0) {
        ((float4*)out)[r] = sb;    // boxes: out[0..400)
        out[400 + r] = s;          // scores: out[400..500)
    }
    float sa = (sb.z - sb.x) * (sb.w - sb.y);

    int mm = *m; if (mm > cap) mm = cap;

#if defined(USE_ASYNC_LDS)
    __shared__ float4             bx_s[2][BT];
    __shared__ unsigned long long key_s[2][BT];
    const int tid = threadIdx.x;
    long long t = blockIdx.x;
    long long G = gridDim.x;
    int buf = 0;
    if (t * BT < mm) {
        int j = (int)(t * BT) + tid;
        if (j < mm) {
            __builtin_amdgcn_global_load_async_to_lds_b128(GPTR4(&bxc[j]),  LPTR4(&bx_s[0][tid]),  0, 0);
            __builtin_amdgcn_global_load_async_to_lds_b64 (GPTR2(&keyc[j]), LPTR2(&key_s[0][tid]), 0, 0);
        }
    }
    for (; t * BT < mm; t += G, buf ^= 1) {
        long long tn = t + G;
        if (tn * BT < mm) {
            int jn = (int)(tn * BT) + tid;
            if (jn < mm) {
                __builtin_amdgcn_global_load_async_to_lds_b128(GPTR4(&bxc[jn]),  LPTR4(&bx_s[buf ^ 1][tid]),  0, 0);
                __builtin_amdgcn_global_load_async_to_lds_b64 (GPTR2(&keyc[jn]), LPTR2(&key_s[buf ^ 1][tid]), 0, 0);
            }
            __builtin_amdgcn_s_wait_asynccnt(2);  // current tile's 2 ops retired (in-order)
        } else {
            __builtin_amdgcn_s_wait_asynccnt(0);
        }
        int j = (int)(t * BT) + tid;
        if (j < mm) {
            unsigned long long kj = key_s[buf][tid];
            if (kj != 0ull) {
                float4 b = bx_s[buf][tid];
                unsigned int ji = 0xFFFFFFFFu - (unsigned int)(kj & 0xFFFFFFFFull);
                if (ji == sel_i || iou_suppress(b, sb, sa)) keyc[j] = 0ull;
            }
        }
    }
#else
    int stride = gridDim.x * blockDim.x;
    for (int j = blockIdx.x * blockDim.x + threadIdx.x; j < mm; j += stride) {
        unsigned long long kj = keyc[j];
        if (kj == 0ull) continue;
        float4 b = bxc[j];
        unsigned int ji = 0xFFFFFFFFu - (unsigned int)(kj & 0xFFFFFFFFull);
        if (ji == sel_i || iou_suppress(b, sb, sa)) keyc[j] = 0ull;
    }
#endif
}

// ---------------- init: zero per-round keys, survivor counter, and d_out ----------------
__global__ __launch_bounds__(512) void rpn_init(unsigned long long* keys, int* m, float* out) {
    int t = threadIdx.x;
    if (t < 128) keys[t] = 0ull;
    if (t == 128) *m = 0;
    for (int i = t; i < 500; i += 512) out[i] = 0.0f;
}

// ---------------- compact: sigmoid(l0-l1) > 0.7 -> append (key, box) ----------------
__global__ __launch_bounds__(BT) void rpn_compact(const float* __restrict__ cls,
                                                  const float* __restrict__ prop,
                                                  unsigned long long* __restrict__ keyc,
                                                  float4* __restrict__ bxc,
                                                  int* __restrict__ m, int cap) {
    const float2* __restrict__ cls2 = (const float2*)cls;
    const float4* __restrict__ prop4 = (const float4*)prop;
    int stride = gridDim.x * blockDim.x;
    for (int i = blockIdx.x * blockDim.x + threadIdx.x; i < N_TOTAL; i += stride) {
        if (i + stride < N_TOTAL) {
            __builtin_prefetch(cls2 + i + stride, 0, 0);   // global_prefetch_b8
        }
        float2 l = cls2[i];
        // softmax([l0,l1])[0] == sigmoid(l0 - l1)
        float p = 1.0f / (1.0f + __expf(l.y - l.x));
        if (p > CLS_TH) {
            int pos = atomicAdd(m, 1);
            if (pos < cap) {
                keyc[pos] = pack_key(p, (unsigned int)i);
                bxc[pos]  = prop4[i];
            }
        }
    }
}

// ---------------- argmax over survivor keys -> keys[r] (deterministic tie-break) ------
__global__ __launch_bounds__(BT) void rpn_argmax(const unsigned long long* __restrict__ keyc,
                                                 const int* __restrict__ m, int cap,
                                                 unsigned long long* __restrict__ keys, int r) {
    int mm = *m; if (mm > cap) mm = cap;
    unsigned long long best = 0ull;
    int stride = gridDim.x * blockDim.x;
    for (int j = blockIdx.x * blockDim.x + threadIdx.x; j < mm; j += stride) {
        unsigned long long k = keyc[j];   // dead entries are 0 -> no branch needed
        best = (k > best) ? k : best;
    }
    // wave32 reduction
    for (int off = 16; off > 0; off >>= 1) {
        unsigned long long o = (unsigned long long)__shfl_down(best, off);
        best = (o > best) ? o : best;
    }
    __shared__ unsigned long long wred[BT / 32];
    int lane = threadIdx.x & 31, wid = threadIdx.x >> 5;
    if (lane == 0) wred[wid] = best;
    __syncthreads();
    if (threadIdx.x == 0) {
        unsigned long long b = wred[0];
        for (int w = 1; w < BT / 32; w++) b = (wred[w] > b) ? wred[w] : b;
        if (b) atomicMax(&keys[r], b);
    }
}

extern "C" void kernel_launch(void* const* d_in, const int* in_sizes, int n_in,
                              void* d_out, int out_size, void* d_ws, size_t ws_size,
                              hipStream_t stream) {
    const float* prop = (const float*)d_in[0];   // proposals [B,H,W,A,4]
    const float* cls  = (const float*)d_in[1];   // cls_score [B,H,W,A*2]
    float* out = (float*)d_out;                  // 400 box floats + 100 scores

    char* ws = (char*)d_ws;
    unsigned long long* keys = (unsigned long long*)ws;          // 128 * 8 B
    int* m = (int*)(ws + 1024);
    size_t avail = (ws_size > 4096) ? (ws_size - 4096) : 0;
    long long cap_ll = (long long)(avail / 24);                  // 16 B box + 8 B key
    if (cap_ll > N_TOTAL) cap_ll = N_TOTAL;
    int cap = (int)cap_ll;
    float4* bxc = (float4*)(ws + 4096);                          // 16B-aligned
    unsigned long long* keyc = (unsigned long long*)(ws + 4096 + (size_t)cap * 16);

    rpn_init<<<1, 512, 0, stream>>>(keys, m, out);
    rpn_compact<<<4096, BT, 0, stream>>>(cls, prop, keyc, bxc, m, cap);
    for (int r = 0; r < ROUNDS; r++) {
        rpn_argmax<<<2048, BT, 0, stream>>>(keyc, m, cap, keys, r);
        rpn_suppress<<<2048, BT, 0, stream>>>(prop, keys, r, keyc, bxc, m, cap, out);
    }
}